// FeatureLayer_4002909520030
// MI455X (gfx1250) — compile-verified
//
#include <hip/hip_runtime.h>
#include <cstddef>

typedef float v2f __attribute__((ext_vector_type(2)));
typedef float v8f __attribute__((ext_vector_type(8)));

#define NSEG_NODE 500000
#define NSEG_EDGE 1000000

// tanh via native v_exp_f32 / v_rcp_f32 (no libcalls):
// tanh(x) = 1 - 2/(exp(2x)+1);  exp(2x) = 2^(x * 2*log2(e))
__device__ __forceinline__ float fast_tanh(float x) {
    float e = __builtin_amdgcn_exp2f(x * 2.885390081777927f);
    return 1.0f - 2.0f * __builtin_amdgcn_rcpf(e + 1.0f);
}

__device__ __forceinline__ float gelu_tanh(float x) {
    float x3 = x * x * x;
    float t = fast_tanh(0.7978845608028654f * (x + 0.044715f * x3));
    return 0.5f * x * (1.0f + t);
}

// Guaranteed native f32 atomic add (no CAS loop).
__device__ __forceinline__ void atomic_add_f32(float* p, float v) {
    asm volatile("global_atomic_add_f32 %0, %1, off" :: "v"(p), "v"(v) : "memory");
}

// D = A(16x4 f32) * B(4x16 f32) + C(16x16 f32)
__device__ __forceinline__ v8f wmma4(v2f a, v2f b, v8f c) {
    return __builtin_amdgcn_wmma_f32_16x16x4_f32(false, a, false, b, (short)0, c,
                                                 false, false);
}

// ---------------------------------------------------------------------------
// Phase 1, nodes: embed -> MLP(48->32->32->32, gelu) -> atomic scatter-sum
// One wave handles 16 nodes per iteration.
// A-operand layout (f32 16x4): lane L holds row M=L&15, K = vgpr + 2*(L>>4).
// B-operand layout (f32 4x16): lane L holds col N=L&15, K = vgpr + 2*(L>>4).
// C/D layout: vgpr v -> M = v + 8*(L>>4), N = (L&15) + 16*tile.
// ---------------------------------------------------------------------------
__global__ __launch_bounds__(256) void node_phase1(
    const int* __restrict__ nf, const int* __restrict__ nidx,
    const float* __restrict__ atom_emb, const float* __restrict__ conn_emb,
    const float* __restrict__ fmchg_emb, const float* __restrict__ ringcon_emb,
    const float* __restrict__ minring_emb,
    const float* __restrict__ W1, const float* __restrict__ B1,
    const float* __restrict__ W2, const float* __restrict__ B2,
    const float* __restrict__ W3, const float* __restrict__ B3,
    float* __restrict__ ssum, float* __restrict__ scnt, int n_nodes)
{
    __shared__ float sW1[1536];      // [kc=12][tile=2][lane=32][j=2]
    __shared__ float sW2[1024];      // [kc=8 ][tile=2][lane=32][j=2]
    __shared__ float sW3[1024];
    __shared__ float sB[96];
    __shared__ float sBuf[8][768];   // per-wave: raw 16x48, reused as h 16x32
    __shared__ int   sIdx[8][16];

    const int tid = threadIdx.x;
    // Pre-permute weights into WMMA-B operand order.
    for (int e = tid; e < 1536; e += 256) {
        int j = e & 1; int r = e >> 1; int ln = r & 31; r >>= 5;
        int t = r & 1; int kc = r >> 1;
        int k = 4 * kc + 2 * (ln >> 4) + j;
        int n = 16 * t + (ln & 15);
        sW1[e] = W1[k * 32 + n];
    }
    for (int e = tid; e < 1024; e += 256) {
        int j = e & 1; int r = e >> 1; int ln = r & 31; r >>= 5;
        int t = r & 1; int kc = r >> 1;
        int k = 4 * kc + 2 * (ln >> 4) + j;
        int n = 16 * t + (ln & 15);
        sW2[e] = W2[k * 32 + n];
        sW3[e] = W3[k * 32 + n];
    }
    if (tid < 32) { sB[tid] = B1[tid]; sB[32 + tid] = B2[tid]; sB[64 + tid] = B3[tid]; }
    __syncthreads();

    const int wid = tid >> 5, lane = tid & 31;
    const int m = lane & 15, half = lane >> 4;
    float* buf  = sBuf[wid];
    int*   widx = sIdx[wid];

    const int nChunks = n_nodes >> 4;
    const int stride  = gridDim.x * 8;
    for (int ch = blockIdx.x * 8 + wid; ch < nChunks; ch += stride) {
        const int node = ch * 16 + m;
        const int* f = nf + (size_t)node * 5;
        if (half == 0) {
            float4* dst = (float4*)(buf + m * 48);
            const float4* s0 = (const float4*)(atom_emb + f[0] * 16);
            dst[0] = s0[0]; dst[1] = s0[1]; dst[2] = s0[2]; dst[3] = s0[3];
            const float4* s1 = (const float4*)(conn_emb + f[1] * 8);
            dst[4] = s1[0]; dst[5] = s1[1];
            widx[m] = nidx[node];
        } else {
            int fm = f[2] + 2; fm = fm < 0 ? 0 : (fm > 4 ? 4 : fm);
            float4* dst = (float4*)(buf + m * 48 + 24);
            const float4* s0 = (const float4*)(fmchg_emb + fm * 8);
            dst[0] = s0[0]; dst[1] = s0[1];
            const float4* s1 = (const float4*)(ringcon_emb + f[3] * 8);
            dst[2] = s1[0]; dst[3] = s1[1];
            const float4* s2 = (const float4*)(minring_emb + f[4] * 8);
            dst[4] = s2[0]; dst[5] = s2[1];
        }
        __builtin_amdgcn_wave_barrier();

        // ---- layer 1: [16x48] x [48x32] ----
        v8f a0 = {}, a1 = {};
        #pragma unroll
        for (int kc = 0; kc < 12; ++kc) {
            v2f A   = *(const v2f*)(buf + m * 48 + 4 * kc + 2 * half);
            v2f Bt0 = *(const v2f*)(sW1 + ((kc * 2 + 0) * 32 + lane) * 2);
            v2f Bt1 = *(const v2f*)(sW1 + ((kc * 2 + 1) * 32 + lane) * 2);
            a0 = wmma4(A, Bt0, a0);
            a1 = wmma4(A, Bt1, a1);
        }
        {
            float bb0 = sB[m], bb1 = sB[m + 16];
            #pragma unroll
            for (int v = 0; v < 8; ++v) {
                a0[v] = gelu_tanh(a0[v] + bb0);
                a1[v] = gelu_tanh(a1[v] + bb1);
            }
        }
        __builtin_amdgcn_wave_barrier();
        #pragma unroll
        for (int v = 0; v < 8; ++v) {
            int rr = v + 8 * half;
            buf[rr * 32 + m]      = a0[v];
            buf[rr * 32 + m + 16] = a1[v];
        }
        __builtin_amdgcn_wave_barrier();

        // ---- layer 2: [16x32] x [32x32] ----
        v8f c0 = {}, c1 = {};
        #pragma unroll
        for (int kc = 0; kc < 8; ++kc) {
            v2f A   = *(const v2f*)(buf + m * 32 + 4 * kc + 2 * half);
            v2f Bt0 = *(const v2f*)(sW2 + ((kc * 2 + 0) * 32 + lane) * 2);
            v2f Bt1 = *(const v2f*)(sW2 + ((kc * 2 + 1) * 32 + lane) * 2);
            c0 = wmma4(A, Bt0, c0);
            c1 = wmma4(A, Bt1, c1);
        }
        {
            float bb0 = sB[32 + m], bb1 = sB[32 + m + 16];
            #pragma unroll
            for (int v = 0; v < 8; ++v) {
                c0[v] = gelu_tanh(c0[v] + bb0);
                c1[v] = gelu_tanh(c1[v] + bb1);
            }
        }
        __builtin_amdgcn_wave_barrier();
        #pragma unroll
        for (int v = 0; v < 8; ++v) {
            int rr = v + 8 * half;
            buf[rr * 32 + m]      = c0[v];
            buf[rr * 32 + m + 16] = c1[v];
        }
        __builtin_amdgcn_wave_barrier();

        // ---- layer 3: [16x32] x [32x32] ----
        v8f d0 = {}, d1 = {};
        #pragma unroll
        for (int kc = 0; kc < 8; ++kc) {
            v2f A   = *(const v2f*)(buf + m * 32 + 4 * kc + 2 * half);
            v2f Bt0 = *(const v2f*)(sW3 + ((kc * 2 + 0) * 32 + lane) * 2);
            v2f Bt1 = *(const v2f*)(sW3 + ((kc * 2 + 1) * 32 + lane) * 2);
            d0 = wmma4(A, Bt0, d0);
            d1 = wmma4(A, Bt1, d1);
        }
        {
            float bb0 = sB[64 + m], bb1 = sB[64 + m + 16];
            #pragma unroll
            for (int v = 0; v < 8; ++v) {
                d0[v] = gelu_tanh(d0[v] + bb0);
                d1[v] = gelu_tanh(d1[v] + bb1);
            }
        }

        // ---- scatter-sum into segments ----
        #pragma unroll
        for (int v = 0; v < 8; ++v) {
            int rr  = v + 8 * half;
            int seg = widx[rr];
            float* p = ssum + (size_t)seg * 32;
            atomic_add_f32(p + m,      d0[v]);
            atomic_add_f32(p + m + 16, d1[v]);
        }
        if (half == 0) atomic_add_f32(scnt + widx[m], 1.0f);
        __builtin_amdgcn_wave_barrier();
    }
}

// ---------------------------------------------------------------------------
// Phase 1, edges: embed(16) -> MLP(16->32->32->32, gelu) -> atomic scatter-sum
// ---------------------------------------------------------------------------
__global__ __launch_bounds__(256) void edge_phase1(
    const int* __restrict__ ef, const int* __restrict__ eidx,
    const float* __restrict__ bondring_emb, const float* __restrict__ bondorder_emb,
    const float* __restrict__ W1, const float* __restrict__ B1,
    const float* __restrict__ W2, const float* __restrict__ B2,
    const float* __restrict__ W3, const float* __restrict__ B3,
    float* __restrict__ ssum, float* __restrict__ scnt, int n_edges)
{
    __shared__ float sW1[512];       // [kc=4][tile=2][lane=32][j=2]
    __shared__ float sW2[1024];
    __shared__ float sW3[1024];
    __shared__ float sB[96];
    __shared__ float sBuf[8][512];   // raw 16x16, reused as h 16x32
    __shared__ int   sIdx[8][16];

    const int tid = threadIdx.x;
    for (int e = tid; e < 512; e += 256) {
        int j = e & 1; int r = e >> 1; int ln = r & 31; r >>= 5;
        int t = r & 1; int kc = r >> 1;
        int k = 4 * kc + 2 * (ln >> 4) + j;
        int n = 16 * t + (ln & 15);
        sW1[e] = W1[k * 32 + n];
    }
    for (int e = tid; e < 1024; e += 256) {
        int j = e & 1; int r = e >> 1; int ln = r & 31; r >>= 5;
        int t = r & 1; int kc = r >> 1;
        int k = 4 * kc + 2 * (ln >> 4) + j;
        int n = 16 * t + (ln & 15);
        sW2[e] = W2[k * 32 + n];
        sW3[e] = W3[k * 32 + n];
    }
    if (tid < 32) { sB[tid] = B1[tid]; sB[32 + tid] = B2[tid]; sB[64 + tid] = B3[tid]; }
    __syncthreads();

    const int wid = tid >> 5, lane = tid & 31;
    const int m = lane & 15, half = lane >> 4;
    float* buf  = sBuf[wid];
    int*   widx = sIdx[wid];

    const int nChunks = n_edges >> 4;
    const int stride  = gridDim.x * 8;
    for (int ch = blockIdx.x * 8 + wid; ch < nChunks; ch += stride) {
        const int edge = ch * 16 + m;
        if (half == 0) {
            float4* dst = (float4*)(buf + m * 16);
            const float4* s0 = (const float4*)(bondring_emb + ef[(size_t)edge * 2] * 8);
            dst[0] = s0[0]; dst[1] = s0[1];
            widx[m] = eidx[edge];
        } else {
            float4* dst = (float4*)(buf + m * 16 + 8);
            const float4* s0 = (const float4*)(bondorder_emb + ef[(size_t)edge * 2 + 1] * 8);
            dst[0] = s0[0]; dst[1] = s0[1];
        }
        __builtin_amdgcn_wave_barrier();

        // ---- layer 1: [16x16] x [16x32] ----
        v8f a0 = {}, a1 = {};
        #pragma unroll
        for (int kc = 0; kc < 4; ++kc) {
            v2f A   = *(const v2f*)(buf + m * 16 + 4 * kc + 2 * half);
            v2f Bt0 = *(const v2f*)(sW1 + ((kc * 2 + 0) * 32 + lane) * 2);
            v2f Bt1 = *(const v2f*)(sW1 + ((kc * 2 + 1) * 32 + lane) * 2);
            a0 = wmma4(A, Bt0, a0);
            a1 = wmma4(A, Bt1, a1);
        }
        {
            float bb0 = sB[m], bb1 = sB[m + 16];
            #pragma unroll
            for (int v = 0; v < 8; ++v) {
                a0[v] = gelu_tanh(a0[v] + bb0);
                a1[v] = gelu_tanh(a1[v] + bb1);
            }
        }
        __builtin_amdgcn_wave_barrier();
        #pragma unroll
        for (int v = 0; v < 8; ++v) {
            int rr = v + 8 * half;
            buf[rr * 32 + m]      = a0[v];
            buf[rr * 32 + m + 16] = a1[v];
        }
        __builtin_amdgcn_wave_barrier();

        // ---- layer 2 ----
        v8f c0 = {}, c1 = {};
        #pragma unroll
        for (int kc = 0; kc < 8; ++kc) {
            v2f A   = *(const v2f*)(buf + m * 32 + 4 * kc + 2 * half);
            v2f Bt0 = *(const v2f*)(sW2 + ((kc * 2 + 0) * 32 + lane) * 2);
            v2f Bt1 = *(const v2f*)(sW2 + ((kc * 2 + 1) * 32 + lane) * 2);
            c0 = wmma4(A, Bt0, c0);
            c1 = wmma4(A, Bt1, c1);
        }
        {
            float bb0 = sB[32 + m], bb1 = sB[32 + m + 16];
            #pragma unroll
            for (int v = 0; v < 8; ++v) {
                c0[v] = gelu_tanh(c0[v] + bb0);
                c1[v] = gelu_tanh(c1[v] + bb1);
            }
        }
        __builtin_amdgcn_wave_barrier();
        #pragma unroll
        for (int v = 0; v < 8; ++v) {
            int rr = v + 8 * half;
            buf[rr * 32 + m]      = c0[v];
            buf[rr * 32 + m + 16] = c1[v];
        }
        __builtin_amdgcn_wave_barrier();

        // ---- layer 3 ----
        v8f d0 = {}, d1 = {};
        #pragma unroll
        for (int kc = 0; kc < 8; ++kc) {
            v2f A   = *(const v2f*)(buf + m * 32 + 4 * kc + 2 * half);
            v2f Bt0 = *(const v2f*)(sW3 + ((kc * 2 + 0) * 32 + lane) * 2);
            v2f Bt1 = *(const v2f*)(sW3 + ((kc * 2 + 1) * 32 + lane) * 2);
            d0 = wmma4(A, Bt0, d0);
            d1 = wmma4(A, Bt1, d1);
        }
        {
            float bb0 = sB[64 + m], bb1 = sB[64 + m + 16];
            #pragma unroll
            for (int v = 0; v < 8; ++v) {
                d0[v] = gelu_tanh(d0[v] + bb0);
                d1[v] = gelu_tanh(d1[v] + bb1);
            }
        }

        #pragma unroll
        for (int v = 0; v < 8; ++v) {
            int rr  = v + 8 * half;
            int seg = widx[rr];
            float* p = ssum + (size_t)seg * 32;
            atomic_add_f32(p + m,      d0[v]);
            atomic_add_f32(p + m + 16, d1[v]);
        }
        if (half == 0) atomic_add_f32(scnt + widx[m], 1.0f);
        __builtin_amdgcn_wave_barrier();
    }
}

// ---------------------------------------------------------------------------
// Phase 2: out[i][:] = seg_sum[idx[i]][:] / max(cnt[idx[i]], 1)
// One thread = one float4 (b128 streaming).
// ---------------------------------------------------------------------------
__global__ __launch_bounds__(256) void gather_mean(
    const float* __restrict__ ssum, const float* __restrict__ scnt,
    const int* __restrict__ idx, float* __restrict__ out, int n)
{
    int t = blockIdx.x * 256 + threadIdx.x;
    if (t >= n * 8) return;
    int i = t >> 3, q = t & 7;
    int seg = idx[i];
    float c = scnt[seg];
    float scale = 1.0f / fmaxf(c, 1.0f);
    float4 v = ((const float4*)(ssum + (size_t)seg * 32))[q];
    float4 r;
    r.x = v.x * scale; r.y = v.y * scale; r.z = v.z * scale; r.w = v.w * scale;
    ((float4*)(out + (size_t)i * 32))[q] = r;
}

extern "C" void kernel_launch(void* const* d_in, const int* in_sizes, int n_in,
                              void* d_out, int out_size, void* d_ws, size_t ws_size,
                              hipStream_t stream) {
    const int*   nf       = (const int*)  d_in[0];   // [N,5] int32
    const int*   ef       = (const int*)  d_in[1];   // [E,2] int32
    const int*   nidx     = (const int*)  d_in[2];
    const int*   eidx     = (const int*)  d_in[3];
    const float* atom_e   = (const float*)d_in[4];
    const float* conn_e   = (const float*)d_in[5];
    const float* fmchg_e  = (const float*)d_in[6];
    const float* ringc_e  = (const float*)d_in[7];
    const float* minr_e   = (const float*)d_in[8];
    const float* bring_e  = (const float*)d_in[9];
    const float* border_e = (const float*)d_in[10];
    const float* nW1 = (const float*)d_in[11]; const float* nb1 = (const float*)d_in[12];
    const float* nW2 = (const float*)d_in[13]; const float* nb2 = (const float*)d_in[14];
    const float* nW3 = (const float*)d_in[15]; const float* nb3 = (const float*)d_in[16];
    const float* eW1 = (const float*)d_in[17]; const float* eb1 = (const float*)d_in[18];
    const float* eW2 = (const float*)d_in[19]; const float* eb2 = (const float*)d_in[20];
    const float* eW3 = (const float*)d_in[21]; const float* eb3 = (const float*)d_in[22];

    const int n_nodes = in_sizes[0] / 5;
    const int n_edges = in_sizes[1] / 2;

    // Workspace: node seg sums [NSEG_NODE*32], node cnt, edge seg sums, edge cnt
    float* nsum = (float*)d_ws;
    float* ncnt = nsum + (size_t)NSEG_NODE * 32;
    float* esum = ncnt + NSEG_NODE;
    float* ecnt = esum + (size_t)NSEG_EDGE * 32;
    size_t ws_floats = (size_t)NSEG_NODE * 33 + (size_t)NSEG_EDGE * 33;
    (void)hipMemsetAsync(d_ws, 0, ws_floats * sizeof(float), stream);

    float* x_h = (float*)d_out;                       // [N,32]
    float* e_h = x_h + (size_t)n_nodes * 32;          // [E,32]

    node_phase1<<<1024, 256, 0, stream>>>(
        nf, nidx, atom_e, conn_e, fmchg_e, ringc_e, minr_e,
        nW1, nb1, nW2, nb2, nW3, nb3, nsum, ncnt, n_nodes);

    edge_phase1<<<1024, 256, 0, stream>>>(
        ef, eidx, bring_e, border_e,
        eW1, eb1, eW2, eb2, eW3, eb3, esum, ecnt, n_edges);

    int nthr = n_nodes * 8;
    gather_mean<<<(nthr + 255) / 256, 256, 0, stream>>>(nsum, ncnt, nidx, x_h, n_nodes);
    int ethr = n_edges * 8;
    gather_mean<<<(ethr + 255) / 256, 256, 0, stream>>>(esum, ecnt, eidx, e_h, n_edges);
}